// Effi_MVS_plus_43069932044812
// MI455X (gfx1250) — compile-verified
//
#include <hip/hip_runtime.h>

// Fixed problem shape from setup_inputs()
#define NV 5      // views
#define NB 2      // batch
#define NC 32     // channels
#define NH 128    // height
#define NW 160    // width
#define ND 32     // depth hypotheses  (== wave32 lanes!)
#define NG 8      // groups
#define HWSZ (NH * NW)

typedef float v2f __attribute__((ext_vector_type(2)));
typedef float v8f __attribute__((ext_vector_type(8)));

__device__ __forceinline__ float wave_max32(float v) {
#pragma unroll
  for (int m = 16; m >= 1; m >>= 1) v = fmaxf(v, __shfl_xor(v, m, 32));
  return v;
}
__device__ __forceinline__ float wave_sum32(float v) {
#pragma unroll
  for (int m = 16; m >= 1; m >>= 1) v += __shfl_xor(v, m, 32);
  return v;
}
__device__ __forceinline__ int iclamp(int v, int lo, int hi) {
  return v < lo ? lo : (v > hi ? hi : v);
}

// make_proj: given p[2][4][4] (extrinsic-like E, intrinsic-like K),
// out = E with rows 0..2 (all 4 cols) replaced by K[:3,:3] @ E[:3,:4]
__device__ void make_proj(const float* p, float* out) {
  const float* e = p;
  const float* k = p + 16;
#pragma unroll
  for (int i = 0; i < 16; ++i) out[i] = e[i];
#pragma unroll
  for (int r = 0; r < 3; ++r)
#pragma unroll
    for (int c = 0; c < 4; ++c) {
      float s = 0.f;
#pragma unroll
      for (int kk = 0; kk < 3; ++kk) s += k[r * 4 + kk] * e[kk * 4 + c];
      out[r * 4 + c] = s;
    }
}

// 4x4 inverse, Gauss-Jordan with partial pivoting
__device__ void inv4(const float* A, float* out) {
  float a[4][8];
  for (int i = 0; i < 4; ++i)
    for (int j = 0; j < 4; ++j) {
      a[i][j] = A[i * 4 + j];
      a[i][4 + j] = (i == j) ? 1.f : 0.f;
    }
  for (int col = 0; col < 4; ++col) {
    int piv = col;
    float best = fabsf(a[col][col]);
    for (int r = col + 1; r < 4; ++r) {
      float v = fabsf(a[r][col]);
      if (v > best) { best = v; piv = r; }
    }
    if (piv != col)
      for (int j = 0; j < 8; ++j) {
        float t = a[col][j]; a[col][j] = a[piv][j]; a[piv][j] = t;
      }
    float d = 1.f / a[col][col];
    for (int j = 0; j < 8; ++j) a[col][j] *= d;
    for (int r = 0; r < 4; ++r)
      if (r != col) {
        float f = a[r][col];
        for (int j = 0; j < 8; ++j) a[r][j] -= f * a[col][j];
      }
  }
  for (int i = 0; i < 4; ++i)
    for (int j = 0; j < 4; ++j) out[i * 4 + j] = a[i][4 + j];
}

// Kernel 0: per (b, v=1..4) relative projection rows 0..2 (rot 3x3 | trans 3)
__global__ void proj_prep_kernel(const float* __restrict__ pm,
                                 float* __restrict__ projrel) {
  int t = threadIdx.x;
  if (t >= NB * (NV - 1)) return;
  int b = t / (NV - 1);
  int v = t % (NV - 1) + 1;
  float refm[16], srcm[16], refinv[16];
  make_proj(pm + (size_t)(b * NV + 0) * 32, refm);
  make_proj(pm + (size_t)(b * NV + v) * 32, srcm);
  inv4(refm, refinv);
  float* o = projrel + t * 12;
  for (int r = 0; r < 3; ++r)
    for (int c = 0; c < 4; ++c) {
      float s = 0.f;
      for (int k = 0; k < 4; ++k) s += srcm[r * 4 + k] * refinv[k * 4 + c];
      o[r * 4 + c] = s;
    }
}

// Main fused kernel: one wave32 per pixel.
// WMMA layout (V_WMMA_F32_16X16X4_F32):
//   lane L: n = L&15, hi = L>>4
//   A_g[M,k]: lane samples depth d = hd*16 + n, channels 4g+2*hi+{0,1}
//   B_g[k,n]: ref channels, nonzero only in column n==g  -> chained over g,
//   accumulator D[r] (lane L) = 4*sim[d = hd*16 + 8*hi + r, group n]
__global__ __launch_bounds__(32) void mvs_fused_kernel(
    const float* __restrict__ feat, const float* __restrict__ dvals,
    const float* __restrict__ projrel, const float* __restrict__ pw_w,
    const float* __restrict__ pw_b, float* __restrict__ out_depth,
    float* __restrict__ out_conf, float* __restrict__ out_vw,
    float* __restrict__ out_reg) {
  __shared__ float simbuf[ND][NG + 1];  // +1 pad vs bank conflicts

  int pix = blockIdx.x;  // 0 .. NB*HWSZ-1
  int b = pix / HWSZ;
  int hw = pix - b * HWSZ;
  int h = hw / NW;
  int w = hw - h * NW;
  int lane = threadIdx.x;
  int n = lane & 15;
  int hi = lane >> 4;

  // Reference-feature values for this lane's B-fragment (group n, if n<8)
  const float* refb = feat + (size_t)b * NC * HWSZ + hw;
  int gch = 4 * (n < NG ? n : NG - 1) + 2 * hi;  // clamp for lanes n>=8 (masked)
  float bs0 = refb[(size_t)gch * HWSZ];
  float bs1 = refb[(size_t)(gch + 1) * HWSZ];

  float simsum[NG];  // accumulators in lane=depth layout
#pragma unroll
  for (int g = 0; g < NG; ++g) simsum[g] = 0.f;
  float wsum = 0.f;
  float pww = pw_w[0], pwb = pw_b[0];

  for (int v = 1; v < NV; ++v) {
    const float* M = projrel + (size_t)(b * (NV - 1) + (v - 1)) * 12;
    float r00 = M[0], r01 = M[1], r02 = M[2], tx = M[3];
    float r10 = M[4], r11 = M[5], r12 = M[6], ty = M[7];
    float r20 = M[8], r21 = M[9], r22 = M[10], tz = M[11];
    float fx = (float)w, fy = (float)h;
    float rx = r00 * fx + r01 * fy + r02;
    float ry = r10 * fx + r11 * fy + r12;
    float rz = r20 * fx + r21 * fy + r22;
    const float* srcb = feat + (size_t)(v * NB + b) * NC * HWSZ;

    v8f acc[2];
#pragma unroll
    for (int hd = 0; hd < 2; ++hd) {
      int didx = hd * 16 + n;
      float depth = dvals[((size_t)(b * ND + didx)) * HWSZ + hw];
      float px = rx * depth + tx, py = ry * depth + ty, pz = rz * depth + tz;
      float gx = px / pz, gy = py / pz;
      float x0 = floorf(gx), y0 = floorf(gy);
      float wx1 = gx - x0, wx0 = 1.f - wx1;
      float wy1 = gy - y0, wy0 = 1.f - wy1;
      float x1 = x0 + 1.f, y1 = y0 + 1.f;
      bool vx0 = (x0 >= 0.f) && (x0 <= (float)(NW - 1));
      bool vx1 = (x1 >= 0.f) && (x1 <= (float)(NW - 1));
      bool vy0 = (y0 >= 0.f) && (y0 <= (float)(NH - 1));
      bool vy1 = (y1 >= 0.f) && (y1 <= (float)(NH - 1));
      int ix0 = iclamp((int)x0, 0, NW - 1), ix1 = iclamp((int)x1, 0, NW - 1);
      int iy0 = iclamp((int)y0, 0, NH - 1), iy1 = iclamp((int)y1, 0, NH - 1);
      float w00 = wx0 * wy0 * ((vx0 && vy0) ? 1.f : 0.f);
      float w10 = wx1 * wy0 * ((vx1 && vy0) ? 1.f : 0.f);
      float w01 = wx0 * wy1 * ((vx0 && vy1) ? 1.f : 0.f);
      float w11 = wx1 * wy1 * ((vx1 && vy1) ? 1.f : 0.f);
      int o00 = iy0 * NW + ix0, o10 = iy0 * NW + ix1;
      int o01 = iy1 * NW + ix0, o11 = iy1 * NW + ix1;

      v8f cacc = {0.f, 0.f, 0.f, 0.f, 0.f, 0.f, 0.f, 0.f};
#pragma unroll
      for (int g = 0; g < NG; ++g) {
        int c0 = 4 * g + 2 * hi;
        const float* p0 = srcb + (size_t)c0 * HWSZ;
        const float* p1 = p0 + HWSZ;
        float wa0 = w00 * p0[o00] + w10 * p0[o10] + w01 * p0[o01] + w11 * p0[o11];
        float wa1 = w00 * p1[o00] + w10 * p1[o10] + w01 * p1[o01] + w11 * p1[o11];
        v2f Af; Af.x = wa0; Af.y = wa1;
        float msel = (n == g) ? 1.f : 0.f;
        v2f Bf; Bf.x = bs0 * msel; Bf.y = bs1 * msel;
        // chained accumulation: after g-loop, cacc = 4 * sim[., group n]
        cacc = __builtin_amdgcn_wmma_f32_16x16x4_f32(
            false, Af, false, Bf, (short)0, cacc, false, false);
      }
      acc[hd] = cacc;
    }

    // Transpose WMMA layout -> lane=depth layout through LDS (mean over 4 chans)
    if (n < NG) {
#pragma unroll
      for (int hd = 0; hd < 2; ++hd)
#pragma unroll
        for (int r = 0; r < 8; ++r)
          simbuf[hd * 16 + 8 * hi + r][n] = acc[hd][r] * 0.25f;
    }
    __syncthreads();  // single-wave workgroup: S_NOP + LDS ordering

    // lane = depth d
    float sim[NG];
    float mx = -3.0e38f;
#pragma unroll
    for (int g = 0; g < NG; ++g) {
      sim[g] = simbuf[lane][g];
      mx = fmaxf(mx, sim[g]);
    }
    float ex[NG], esum = 0.f;
#pragma unroll
    for (int g = 0; g < NG; ++g) {
      ex[g] = expf(sim[g] - mx);
      esum += ex[g];
    }
    float inv_s = 1.f / esum;
    float ent = 0.f;
#pragma unroll
    for (int g = 0; g < NG; ++g) {
      float sn = ex[g] * inv_s;
      ent -= sn * logf(sn + 1e-7f);
    }
    float xarg = pww * ent + pwb;
    float xmax = wave_max32(xarg);        // max over all 32 depths
    float vw = 1.f / (1.f + expf(-xmax)); // sigmoid monotonic -> max commutes
#pragma unroll
    for (int g = 0; g < NG; ++g) simsum[g] += sim[g] * vw;
    wsum += vw;
    if (lane == 0)
      out_vw[((size_t)(b * (NV - 1) + (v - 1))) * HWSZ + hw] = vw;
    __syncthreads();  // before next view reuses simbuf
  }

  // Final regularization / depth / confidence, lane = depth d
  float rs = 0.f;
#pragma unroll
  for (int g = 0; g < NG; ++g) rs += simsum[g];
  float regv = rs * (1.0f / NG) / (wsum + 1e-6f);
  out_reg[((size_t)(b * ND + lane)) * HWSZ + hw] = regv;

  float m = wave_max32(regv);
  float e = expf(regv - m);
  float s = wave_sum32(e);
  float prob = e / s;
  float dval = dvals[((size_t)(b * ND + lane)) * HWSZ + hw];
  float depth_out = wave_sum32(prob * dval);
  float idxf = wave_sum32(prob * (float)lane);
  int idx = iclamp((int)floorf(idxf), 0, ND - 1);

  float pm1 = __shfl_up(prob, 1, 32);
  if (lane == 0) pm1 = 0.f;
  float pp1 = __shfl_down(prob, 1, 32);
  if (lane >= ND - 1) pp1 = 0.f;
  float pp2 = __shfl_down(prob, 2, 32);
  if (lane >= ND - 2) pp2 = 0.f;
  float sum4 = pm1 + prob + pp1 + pp2;
  float conf = __shfl(sum4, idx, 32);

  if (lane == 0) {
    out_depth[(size_t)b * HWSZ + hw] = depth_out;
    out_conf[(size_t)b * HWSZ + hw] = conf;
  }
}

extern "C" void kernel_launch(void* const* d_in, const int* in_sizes, int n_in,
                              void* d_out, int out_size, void* d_ws,
                              size_t ws_size, hipStream_t stream) {
  const float* feat = (const float*)d_in[0];   // (V,B,C,H,W) f32
  const float* pm = (const float*)d_in[1];     // (B,V,2,4,4) f32
  const float* dvals = (const float*)d_in[2];  // (B,D,H,W) f32
  // d_in[3]=num_depth, d_in[4]=G (hardcoded)
  const float* pw_w = (const float*)d_in[5];
  const float* pw_b = (const float*)d_in[6];

  float* out = (float*)d_out;
  float* out_depth = out;                                       // (B,H,W)
  float* out_conf = out + (size_t)NB * HWSZ;                    // (B,H,W)
  float* out_vw = out + (size_t)2 * NB * HWSZ;                  // (B,4,H,W)
  float* out_reg = out + (size_t)(2 + (NV - 1)) * NB * HWSZ;    // (B,D,H,W)

  float* projrel = (float*)d_ws;  // NB*(NV-1)*12 floats

  proj_prep_kernel<<<1, 32, 0, stream>>>(pm, projrel);
  mvs_fused_kernel<<<dim3(NB * HWSZ), dim3(32), 0, stream>>>(
      feat, dvals, projrel, pw_w, pw_b, out_depth, out_conf, out_vw, out_reg);
}